// RHTRotation_82394652607202
// MI455X (gfx1250) — compile-verified
//
#include <hip/hip_runtime.h>
#include <hip/hip_bf16.h>

// Randomized Hadamard Transform for MI455X (gfx1250, wave32).
// out = FWHT_4096(x * signs) * 4096^-0.5, N=8192 rows, fp32.
//
// Memory-bound (256 MB total traffic, ~11us floor @ 23.3 TB/s) -> single
// pass: one wave per row, whole transform done in LDS + WMMA.
// H4096 = H16 (x) H16 (x) H16 : three identical LDS->LDS passes, each a set
// of 16 tile matmuls C = H16 x B via V_WMMA_F32_16X16X4_F32 (fp32-exact,
// matches the fp32 reference up to summation order).

typedef float v2f __attribute__((ext_vector_type(2)));
typedef float v8f __attribute__((ext_vector_type(8)));

#define RHT_D 4096
#define RHT_N 8192
#define WAVES_PER_BLOCK 2

// H16[i][j] = +1 / -1 by parity of popcount(i & j)
__device__ __forceinline__ float h16(int i, int j) {
    return (__builtin_popcount(i & j) & 1) ? -1.0f : 1.0f;
}

__global__ void __launch_bounds__(32 * WAVES_PER_BLOCK)
rht_wmma_kernel(const float* __restrict__ x,
                const float* __restrict__ signs,
                float* __restrict__ out) {
    extern __shared__ float lds[];

    const int lane = threadIdx.x & 31;
    const int wave = threadIdx.x >> 5;
    const int row  = blockIdx.x * WAVES_PER_BLOCK + wave;
    if (row >= RHT_N) return;

    // Per-wave private LDS: 4096-float in buffer + 4096-float out buffer.
    float* bufA = lds + wave * (2 * RHT_D);
    float* bufB = bufA + RHT_D;

    const int half = lane >> 4;   // 0: lanes 0-15, 1: lanes 16-31
    const int arow = lane & 15;   // matrix row (A) / column (B,C)

    // ---- A-operand: H16 fragments (constant, built from popcount) --------
    // ISA 32-bit 16x4 A layout: lanes 0-15 -> (row, K=0),(row, K=1) in
    // VGPR0/1 ; lanes 16-31 -> (row, K=2),(row, K=3).
    v2f aFrag[4];
#pragma unroll
    for (int q = 0; q < 4; ++q) {
        const int kb = 4 * q + 2 * half;
        aFrag[q].x = h16(arow, kb);
        aFrag[q].y = h16(arow, kb + 1);
    }

    // ---- Phase 1: stream row from HBM into LDS, applying signs -----------
    const float* xr = x + (size_t)row * RHT_D;
#pragma unroll 4
    for (int it = 0; it < 32; ++it) {
        const int off = (it * 32 + lane) * 4;               // coalesced b128
        const float4 v = *(const float4*)(xr + off);
        const float4 s = *(const float4*)(signs + off);
        float4 w;
        w.x = v.x * s.x; w.y = v.y * s.y;
        w.z = v.z * s.z; w.w = v.w * s.w;
        *(float4*)(bufA + off) = w;
    }
    asm volatile("s_wait_dscnt 0x0" ::: "memory");

    // ---- Phase 2: three Hadamard passes (one per tensor axis) ------------
    // Pass reads src as 16x256 (axis-major), computes H16 x tile for each of
    // the 16 column tiles, and stores C(m,n) of tile t at t*256 + n*16 + m,
    // which rotates the (a,b,c) axes so all three passes are identical.
    for (int stage = 0; stage < 3; ++stage) {
        float* __restrict__ src = (stage & 1) ? bufB : bufA;
        float* __restrict__ dst = (stage & 1) ? bufA : bufB;

        for (int t = 0; t < 16; ++t) {
            v8f acc = {};
#pragma unroll
            for (int q = 0; q < 4; ++q) {
                // B-operand 4x16 fp32: rows striped across lanes mirroring A:
                // lanes 0-15 hold rows (4q+0, 4q+1), lanes 16-31 (4q+2, 4q+3).
                const int kb = 4 * q + 2 * half;
                v2f b;
                b.x = src[kb * 256 + t * 16 + arow];
                b.y = src[(kb + 1) * 256 + t * 16 + arow];
                acc = __builtin_amdgcn_wmma_f32_16x16x4_f32(
                    false, aFrag[q], false, b, (short)0, acc, false, false);
            }
            // C layout: lanes 0-15 hold (m=v, n=lane) v=0..7; lanes 16-31
            // hold (m=v+8, n=lane-16). Dest addresses contiguous in m.
            float* dp = dst + t * 256 + arow * 16 + half * 8;
            float4 lo, hi;
            lo.x = acc[0]; lo.y = acc[1]; lo.z = acc[2]; lo.w = acc[3];
            hi.x = acc[4]; hi.y = acc[5]; hi.z = acc[6]; hi.w = acc[7];
            *(float4*)(dp)     = lo;
            *(float4*)(dp + 4) = hi;
        }
        asm volatile("s_wait_dscnt 0x0" ::: "memory");
    }

    // ---- Phase 3: stream result (in bufB after 3 passes) back to HBM -----
    float* outr = out + (size_t)row * RHT_D;
    const float scale = 0.015625f;  // 4096^-0.5 = 1/64, exact
#pragma unroll 4
    for (int it = 0; it < 32; ++it) {
        const int off = (it * 32 + lane) * 4;
        const float4 v = *(const float4*)(bufB + off);
        float4 w;
        w.x = v.x * scale; w.y = v.y * scale;
        w.z = v.z * scale; w.w = v.w * scale;
        *(float4*)(outr + off) = w;
    }
}

extern "C" void kernel_launch(void* const* d_in, const int* in_sizes, int n_in,
                              void* d_out, int out_size, void* d_ws, size_t ws_size,
                              hipStream_t stream) {
    const float* x     = (const float*)d_in[0];
    const float* signs = (const float*)d_in[1];
    float*       out   = (float*)d_out;

    const dim3 block(32 * WAVES_PER_BLOCK);
    const dim3 grid(RHT_N / WAVES_PER_BLOCK);
    const size_t shmem = (size_t)WAVES_PER_BLOCK * 2 * RHT_D * sizeof(float); // 64 KB

    rht_wmma_kernel<<<grid, block, shmem, stream>>>(x, signs, out);
}